// DepthwiseTensorProduct_55465207660823
// MI455X (gfx1250) — compile-verified
//
#include <hip/hip_runtime.h>
#include <math.h>

// ---------------------------------------------------------------------------
// Depthwise (uvu) e3nn tensor product, MI455X / gfx1250.
// Memory-bound (~1 FLOP/byte): data movement via CDNA5 Tensor Data Mover
// (tensor_load_to_lds + s_wait_tensorcnt), compute via unrolled VALU.
// ---------------------------------------------------------------------------

typedef unsigned int v4u __attribute__((ext_vector_type(4)));
typedef int          v4i __attribute__((ext_vector_type(4)));
typedef int          v8i __attribute__((ext_vector_type(8)));

constexpr int kNI    = 15;
constexpr int kDim1  = 480;
constexpr int kDim2  = 9;
constexpr int kNW    = 960;
constexpr int kDOut  = 3136;
constexpr int kTileZ = 16;
constexpr int kThreads = kTileZ * 32;   // one wave32 per sample

// Instruction tables: (l1,l2,l3) paths in reference order.
constexpr int cL1[kNI]  = {0,0,0, 1,1,1,1,1,1, 2,2,2,2,2,2};
constexpr int cL2[kNI]  = {0,1,2, 0,1,1,1,2,2, 0,1,1,2,2,2};
constexpr int cL3[kNI]  = {0,1,2, 1,0,1,2,1,2, 2,1,2,0,1,2};
constexpr int cD1[kNI]  = {1,1,1, 3,3,3,3,3,3, 5,5,5,5,5,5};
constexpr int cD2[kNI]  = {1,3,5, 1,3,3,3,5,5, 1,3,3,5,5,5};
constexpr int cD3[kNI]  = {1,3,5, 3,1,3,5,3,5, 5,3,5,1,3,5};
constexpr int cS1[kNI]  = {0,0,0, 128,128,128,128,128,128, 320,320,320,320,320,320};
constexpr int cS2[kNI]  = {0,1,4, 0,1,1,1,4,4, 0,1,1,4,4,4};
constexpr int cMul[kNI] = {128,128,128, 64,64,64,64,64,64, 32,32,32,32,32,32};
constexpr int cWO[kNI]  = {0,128,256, 384,448,512,576,640,704, 768,800,832,864,896,928};
constexpr int cOO[kNI]  = {0,128,512, 1152,1344,1408,1600,1920,2112, 2432,2592,2688,2848,2880,2976};
constexpr int cCOFF[kNI+1] = {0,1,10,35,44,53,80,125,170,245,270,315,390,415,490,615};
constexpr int cGOFF[kNI+1] = {0,1,4,9,18,21,30,45,54,69,94,109,134,139,154,179};
constexpr int kGTot = 179;   // total G = (d1*d3) entries per sample
constexpr int kCTot = 615;   // total Wigner-3j floats in d_ws
// pw = sqrt(2*l3+1) (mul2 == 1)
constexpr float cPW[kNI] = {1.0f, 1.7320508f, 2.2360680f,
                            1.7320508f, 1.0f, 1.7320508f, 2.2360680f, 1.7320508f, 2.2360680f,
                            2.2360680f, 1.7320508f, 2.2360680f, 1.0f, 1.7320508f, 2.2360680f};

// ---------------------------------------------------------------------------
// Setup kernel: compute real-basis Wigner-3j tensors (e3nn convention, unit
// Frobenius norm) on device in fp64; exact port of the Python reference.
// ---------------------------------------------------------------------------
struct cd { double re, im; };
__device__ __forceinline__ cd cmul(cd a, cd b) {
  cd r; r.re = a.re*b.re - a.im*b.im; r.im = a.re*b.im + a.im*b.re; return r;
}
__device__ double factd(int n) { double r = 1.0; for (int i = 2; i <= n; ++i) r *= (double)i; return r; }

__device__ double su2_cg(int j1, int j2, int j3, int m1, int m2, int m3) {
  if (m3 != m1 + m2) return 0.0;
  double pref = sqrt((double)(2*j3+1) * factd(j3+j1-j2) * factd(j3-j1+j2) *
                     factd(j1+j2-j3) / factd(j1+j2+j3+1));
  pref *= sqrt(factd(j3+m3)*factd(j3-m3)*factd(j1-m1)*factd(j1+m1)*factd(j2-m2)*factd(j2+m2));
  int kmin = 0;
  if (j2-j3-m1 > kmin) kmin = j2-j3-m1;
  if (j1-j3+m2 > kmin) kmin = j1-j3+m2;
  int kmax = j1+j2-j3;
  if (j1-m1 < kmax) kmax = j1-m1;
  if (j2+m2 < kmax) kmax = j2+m2;
  double s = 0.0;
  for (int k = kmin; k <= kmax; ++k) {
    double d = factd(k)*factd(j1+j2-j3-k)*factd(j1-m1-k)*factd(j2+m2-k)*
               factd(j3-j2+m1+k)*factd(j3-j1-m2+k);
    s += ((k & 1) ? -1.0 : 1.0) / d;
  }
  return pref * s;
}

__device__ void qmat(int l, cd q[5][5]) {
  for (int r = 0; r < 5; ++r) for (int c = 0; c < 5; ++c) q[r][c] = {0.0, 0.0};
  const double inv = 1.0 / sqrt(2.0);
  for (int m = -l; m < 0; ++m) {          // q[l+m, l+|m|]=1/sqrt2 ; q[l+m, l-|m|]=-i/sqrt2
    q[l+m][l-m] = { inv, 0.0 };
    q[l+m][l+m] = { 0.0, -inv };
  }
  q[l][l] = { 1.0, 0.0 };
  for (int m = 1; m <= l; ++m) {
    double s = (m & 1) ? -1.0 : 1.0;
    q[l+m][l+m] = { s*inv, 0.0 };
    q[l+m][l-m] = { 0.0, s*inv };
  }
  cd f;                                    // (-i)^l
  if (l == 0) f = { 1.0, 0.0 };
  else if (l == 1) f = { 0.0, -1.0 };
  else f = { -1.0, 0.0 };
  for (int r = 0; r < 5; ++r) for (int c = 0; c < 5; ++c) q[r][c] = cmul(f, q[r][c]);
}

__global__ void w3j_setup_kernel(float* __restrict__ w3j) {
  const int t = threadIdx.x;
  if (blockIdx.x != 0 || t >= kNI) return;
  const int l1 = cL1[t], l2 = cL2[t], l3 = cL3[t];
  const int d1 = 2*l1+1, d2 = 2*l2+1, d3 = 2*l3+1;
  double C[5][5][5];
  for (int i = 0; i < 5; ++i) for (int k = 0; k < 5; ++k) for (int n = 0; n < 5; ++n) C[i][k][n] = 0.0;
  for (int m1 = -l1; m1 <= l1; ++m1)
    for (int m2 = -l2; m2 <= l2; ++m2) {
      int m3 = m1 + m2;
      if (m3 >= -l3 && m3 <= l3) C[l1+m1][l2+m2][l3+m3] = su2_cg(l1, l2, l3, m1, m2, m3);
    }
  cd Q1[5][5], Q2[5][5], Q3[5][5];
  qmat(l1, Q1); qmat(l2, Q2); qmat(l3, Q3);
  double Cr[5][5][5]; double nrm = 0.0;
  for (int j = 0; j < d1; ++j)
    for (int l = 0; l < d2; ++l)
      for (int m = 0; m < d3; ++m) {
        cd acc = { 0.0, 0.0 };
        for (int i = 0; i < d1; ++i)
          for (int k = 0; k < d2; ++k)
            for (int n = 0; n < d3; ++n) {
              double c = C[i][k][n];
              if (c == 0.0) continue;
              cd q3c = { Q3[n][m].re, -Q3[n][m].im };
              cd p = cmul(Q1[i][j], Q2[k][l]);
              p = cmul(p, q3c);
              acc.re += p.re * c; acc.im += p.im * c;
            }
        Cr[j][l][m] = acc.re;
        nrm += acc.re * acc.re;
      }
  const double inv = 1.0 / sqrt(nrm);
  float* dst = w3j + cCOFF[t];
  for (int j = 0; j < d1; ++j)
    for (int l = 0; l < d2; ++l)
      for (int m = 0; m < d3; ++m)
        dst[(j*d2 + l)*d3 + m] = (float)(Cr[j][l][m] * inv);
}

// ---------------------------------------------------------------------------
// TDM tile load: global [rows x row_elems] f32 (row stride stride_elems) -> LDS.
// Unconditional: if the builtin is absent/mismatched we WANT a compile error
// (compile-only loop; a silent fallback would hide the signature mismatch).
// ---------------------------------------------------------------------------
__device__ __forceinline__ unsigned lds_off(const void* p) {
  // flat LDS aperture: low 32 bits carry the workgroup-relative LDS byte offset
  return (unsigned)(unsigned long long)p;
}

__device__ __forceinline__ void tdm_load_2d_f32(unsigned lds_byte, const float* gptr,
                                                unsigned row_elems, unsigned rows,
                                                unsigned stride_elems) {
  unsigned long long ga = (unsigned long long)(const void*)gptr;
  v4u g0;
  g0.x = 1u;                                                   // count=1
  g0.y = lds_byte;                                             // lds_addr
  g0.z = (unsigned)(ga & 0xFFFFFFFFull);                       // global_addr[31:0]
  g0.w = (unsigned)((ga >> 32) & 0x1FFFFFFull) | (2u << 30);   // addr[56:32] | type=2
  v8i g1;
  g1[0] = (int)(2u << 16);                                     // data_size = 4B
  g1[1] = (int)((row_elems & 0xFFFFu) << 16);                  // tensor_dim0[15:0]
  g1[2] = (int)(((row_elems >> 16) & 0xFFFFu) | ((rows & 0xFFFFu) << 16)); // dim0 hi | dim1 lo
  g1[3] = (int)(((rows >> 16) & 0xFFFFu) | ((row_elems & 0xFFFFu) << 16)); // dim1 hi | tile_dim0
  g1[4] = (int)(rows & 0xFFFFu);                               // tile_dim1 | tile_dim2=0
  g1[5] = (int)stride_elems;                                   // tensor_dim0_stride[31:0]
  g1[6] = 0;
  g1[7] = 0;
  v4i g2 = {0, 0, 0, 0};
  v4i g3 = {0, 0, 0, 0};
#if __clang_major__ >= 23
  v8i g4 = {0, 0, 0, 0, 0, 0, 0, 0};
  __builtin_amdgcn_tensor_load_to_lds(g0, g1, g2, g3, g4, 0);
#else
  __builtin_amdgcn_tensor_load_to_lds(g0, g1, g2, g3, 0);
#endif
}

// ---------------------------------------------------------------------------
// Per-instruction fully unrolled compute: out[u,k] = pw * w[u] * sum_i x[u,i]*G[i,k]
// ---------------------------------------------------------------------------
template <int I>
__device__ __forceinline__ void run_instr(const float* __restrict__ gx,
                                          const float* __restrict__ gw,
                                          const float* __restrict__ gG,
                                          float* __restrict__ outz, int lane) {
  constexpr int d1 = cD1[I], d3 = cD3[I];
  constexpr int reps = cMul[I] / 32;
#pragma unroll
  for (int r = 0; r < reps; ++r) {
    const int u = r * 32 + lane;
    const float wt = gw[cWO[I] + u] * cPW[I];
    float xv[d1];
#pragma unroll
    for (int i = 0; i < d1; ++i) xv[i] = gx[cS1[I] + u * d1 + i];
#pragma unroll
    for (int k = 0; k < d3; ++k) {
      float acc = 0.0f;
#pragma unroll
      for (int i = 0; i < d1; ++i) acc = fmaf(xv[i], gG[cGOFF[I] + i * d3 + k], acc);
      outz[cOO[I] + u * d3 + k] = wt * acc;
    }
  }
}

__global__ __launch_bounds__(kThreads) void tp_kernel(const float* __restrict__ x,
                                                      const float* __restrict__ y,
                                                      const float* __restrict__ w,
                                                      const float* __restrict__ w3j,
                                                      float* __restrict__ out, int Ztot) {
  __shared__ __align__(16) float sx[kTileZ * kDim1];   // 30720 B
  __shared__ __align__(16) float sw[kTileZ * kNW];     // 61440 B
  __shared__ __align__(16) float sy[kTileZ * kDim2];   //   576 B
  __shared__ __align__(16) float sG[kTileZ * kGTot];   // 11456 B

  const int z0 = blockIdx.x * kTileZ;
  int rows = Ztot - z0; if (rows > kTileZ) rows = kTileZ;

  if (threadIdx.x < 32) {   // wave 0 drives the Tensor Data Mover
    tdm_load_2d_f32(lds_off(sx), x + (size_t)z0 * kDim1, kDim1, (unsigned)rows, kDim1);
    tdm_load_2d_f32(lds_off(sw), w + (size_t)z0 * kNW,  kNW,  (unsigned)rows, kNW);
    tdm_load_2d_f32(lds_off(sy), y + (size_t)z0 * kDim2, kDim2, (unsigned)rows, kDim2);
    __builtin_amdgcn_s_wait_tensorcnt(0);
  }
  __syncthreads();

  const int wv = threadIdx.x >> 5;
  const int lane = threadIdx.x & 31;
  const int z = z0 + wv;
  const bool active = (z < Ztot);

  const float* gx = sx + wv * kDim1;
  const float* gw = sw + wv * kNW;
  const float* gy = sy + wv * kDim2;
  float* gG = sG + wv * kGTot;

  if (active) {
    // Cooperative y-contraction: G[i,k] = sum_j C[i,j,k] * y[j]
    for (int e = lane; e < kGTot; e += 32) {
      int c = 0;
#pragma unroll
      for (int t = 1; t < kNI; ++t)
        if (cGOFF[t] <= e) c = t;
      const int loc = e - cGOFF[c];
      const int d3 = cD3[c], d2 = cD2[c];
      const int i = loc / d3;
      const int k = loc - i * d3;
      const float* Cc = w3j + cCOFF[c] + i * d2 * d3 + k;
      const float* yb = gy + cS2[c];
      float acc = 0.0f;
      for (int j = 0; j < d2; ++j) acc = fmaf(Cc[j * d3], yb[j], acc);
      gG[e] = acc;
    }
  }
  __syncthreads();

  if (active) {
    float* outz = out + (size_t)z * kDOut;
    run_instr<0>(gx, gw, gG, outz, lane);
    run_instr<1>(gx, gw, gG, outz, lane);
    run_instr<2>(gx, gw, gG, outz, lane);
    run_instr<3>(gx, gw, gG, outz, lane);
    run_instr<4>(gx, gw, gG, outz, lane);
    run_instr<5>(gx, gw, gG, outz, lane);
    run_instr<6>(gx, gw, gG, outz, lane);
    run_instr<7>(gx, gw, gG, outz, lane);
    run_instr<8>(gx, gw, gG, outz, lane);
    run_instr<9>(gx, gw, gG, outz, lane);
    run_instr<10>(gx, gw, gG, outz, lane);
    run_instr<11>(gx, gw, gG, outz, lane);
    run_instr<12>(gx, gw, gG, outz, lane);
    run_instr<13>(gx, gw, gG, outz, lane);
    run_instr<14>(gx, gw, gG, outz, lane);
  }
}

extern "C" void kernel_launch(void* const* d_in, const int* in_sizes, int n_in,
                              void* d_out, int out_size, void* d_ws, size_t ws_size,
                              hipStream_t stream) {
  const float* x = (const float*)d_in[0];   // [Z, 480]
  const float* y = (const float*)d_in[1];   // [Z, 9]
  const float* w = (const float*)d_in[2];   // [Z, 960]
  float* out = (float*)d_out;               // [Z, 3136]
  float* w3j = (float*)d_ws;                // 615 floats of Wigner-3j data

  const int Z = in_sizes[0] / kDim1;        // 50000
  const int nblocks = (Z + kTileZ - 1) / kTileZ;

  w3j_setup_kernel<<<1, 32, 0, stream>>>(w3j);
  tp_kernel<<<nblocks, kThreads, 0, stream>>>(x, y, w, w3j, out, Z);
}